// CircuitGNNEncoder_65558380806450
// MI455X (gfx1250) — compile-verified
//
#include <hip/hip_runtime.h>
#include <math.h>

// ---------------- problem constants (from reference) ----------------
#define N_NODES  50000
#define N_EDGES  800000
#define N_GRAPHS 512
#define IN_DIM   16
#define EDGE_DIM 8
#define HID      32
#define OUT_DIM  64
#define HEADS    4
#define D2       128          // HEADS*HID
#define NEG_SLOPE 0.2f
#define TPB 256

typedef __attribute__((ext_vector_type(2))) float v2f;
typedef __attribute__((ext_vector_type(8))) float v8f;

static inline int cdiv(int a, int b) { return (a + b - 1) / b; }

// ---------------- float atomic max (IEEE int-punning trick) ----------------
__device__ __forceinline__ void atomicMaxF(float* addr, float val) {
  if (val >= 0.0f) {
    atomicMax((int*)addr, __float_as_int(val));
  } else {
    atomicMin((unsigned int*)addr, __float_as_uint(val));
  }
}

// ---------------- generic fill ----------------
__global__ void fill_kernel(float* __restrict__ p, float v, int n) {
  int i = blockIdx.x * blockDim.x + threadIdx.x;
  if (i < n) p[i] = v;
}

// ---------------- exact-fp32 WMMA GEMM: C[MxNc] = A[MxK] @ W[KxNc] + bias ----
// One wave per 16x16 output tile; K stepped by 4 using V_WMMA_F32_16X16X4_F32.
// K, Nc are compile-time -> fully unrolled, immediate-offset b64/b32 loads.
// A 16x4 layout: lane&15 = M row, (lane>>4)*2+vgpr = K.
// B 4x16 layout: lane&15 = N col, rows 0/2 in vgpr0, rows 1/3 in vgpr1 by half.
// D 16x16: vgpr r -> row r (lanes 0-15) / row r+8 (lanes 16-31), col = lane&15.
// ACT: 0 = none, 1 = relu
template <int K, int Nc, int ACT>
__global__ void wmma_gemm_f32(const float* __restrict__ A,
                              const float* __restrict__ W,
                              const float* __restrict__ bias,
                              float* __restrict__ C, int M) {
  const int lane = threadIdx.x & 31;
  const int wave = threadIdx.x >> 5;
  constexpr int tilesN = Nc >> 4;
  const int tile = blockIdx.x * (blockDim.x >> 5) + wave;
  const int totTiles = (M >> 4) * tilesN;
  if (tile >= totTiles) return;                 // wave-uniform exit: EXEC stays full
  const int tm = tile / tilesN;
  const int tn = tile - tm * tilesN;
  const int row0 = tm << 4;
  const int col0 = tn << 4;

  const int l15  = lane & 15;
  const int koff = (lane >> 4) << 1;            // 0 or 2
  const float* __restrict__ pA = A + (size_t)(row0 + l15) * K + koff;
  const int bcol = col0 + l15;
  const float* __restrict__ pW = W + (size_t)koff * Nc + bcol;

  v8f acc = {};
#pragma unroll
  for (int k = 0; k < K; k += 4) {
    v2f a = *(const v2f*)(pA + k);              // 8B-aligned: even element index
    v2f b;
    b.x = pW[(size_t)k * Nc];
    b.y = pW[(size_t)(k + 1) * Nc];
    acc = __builtin_amdgcn_wmma_f32_16x16x4_f32(false, a, false, b,
                                                (short)0, acc, false, false);
  }

  const int rbase = row0 + ((lane >> 4) << 3);  // +0 or +8
  const float bv = bias[bcol];
#pragma unroll
  for (int r = 0; r < 8; ++r) {
    float v = acc[r] + bv;
    if (ACT == 1) v = v > 0.0f ? v : 0.0f;
    C[(size_t)(rbase + r) * Nc + bcol] = v;
  }
}

// ---------------- edge attention logits + running segment max --------------
// One thread per (edge, head). We [8,H*C] and att [H,C] cached in LDS as f4;
// ea = edge_attr@We recomputed on the fly (never materialized in HBM).
// All gathers are b128 loads against L2-resident node arrays.
template <int H, int C>
__global__ void edge_logits(const float* __restrict__ xs,
                            const float* __restrict__ xd,
                            const int* __restrict__ src,
                            const int* __restrict__ dst,
                            const float* __restrict__ eattr,   // [E,8]
                            const float* __restrict__ We,      // [8,H*C]
                            const float* __restrict__ att,     // [H,C]
                            float* __restrict__ elog,          // [E,H]
                            float* __restrict__ mx,            // [N,H]
                            int nE) {
  constexpr int HC = H * C;
  constexpr int WQ = EDGE_DIM * HC / 4;
  constexpr int AQ = HC / 4;
  __shared__ float4 sWe[WQ];
  __shared__ float4 sAtt[AQ];
  for (int i = threadIdx.x; i < WQ; i += blockDim.x) sWe[i] = ((const float4*)We)[i];
  for (int i = threadIdx.x; i < AQ; i += blockDim.x) sAtt[i] = ((const float4*)att)[i];
  __syncthreads();

  int idx = blockIdx.x * blockDim.x + threadIdx.x;
  if (idx >= nE * H) return;
  int e = idx / H, h = idx - e * H;
  int s = src[e], d = dst[e];

  float4 ea0 = ((const float4*)eattr)[(size_t)e * 2];
  float4 ea1 = ((const float4*)eattr)[(size_t)e * 2 + 1];
  const float ea[EDGE_DIM] = {ea0.x, ea0.y, ea0.z, ea0.w,
                              ea1.x, ea1.y, ea1.z, ea1.w};

  const float4* __restrict__ pxs = (const float4*)(xs + (size_t)s * HC + h * C);
  const float4* __restrict__ pxd = (const float4*)(xd + (size_t)d * HC + h * C);
  const float4* __restrict__ pwe = &sWe[h * (C / 4)];
  const float4* __restrict__ pat = &sAtt[h * (C / 4)];

  float acc = 0.0f;
#pragma unroll
  for (int c4 = 0; c4 < C / 4; ++c4) {
    float4 v = make_float4(0.f, 0.f, 0.f, 0.f);
#pragma unroll
    for (int j = 0; j < EDGE_DIM; ++j) {
      float4 w = pwe[j * (HC / 4) + c4];
      v.x += ea[j] * w.x; v.y += ea[j] * w.y;
      v.z += ea[j] * w.z; v.w += ea[j] * w.w;
    }
    float4 s4 = pxs[c4];
    float4 d4 = pxd[c4];
    float4 a4 = pat[c4];
    float zx = s4.x + d4.x + v.x; zx = zx > 0.f ? zx : NEG_SLOPE * zx;
    float zy = s4.y + d4.y + v.y; zy = zy > 0.f ? zy : NEG_SLOPE * zy;
    float zz = s4.z + d4.z + v.z; zz = zz > 0.f ? zz : NEG_SLOPE * zz;
    float zw = s4.w + d4.w + v.w; zw = zw > 0.f ? zw : NEG_SLOPE * zw;
    acc += zx * a4.x + zy * a4.y + zz * a4.z + zw * a4.w;
  }
  elog[idx] = acc;
  atomicMaxF(&mx[d * H + h], acc);
}

// ---------------- softmax numerator + denominator accumulation -------------
template <int H>
__global__ void edge_softmax_num(const int* __restrict__ dst,
                                 float* __restrict__ elog,
                                 const float* __restrict__ mx,
                                 float* __restrict__ den, int nE) {
  int idx = blockIdx.x * blockDim.x + threadIdx.x;
  if (idx >= nE * H) return;
  int e = idx / H, h = idx - e * H;
  int d = dst[e];
  float p = expf(elog[idx] - mx[d * H + h]);
  elog[idx] = p;
  atomicAdd(&den[d * H + h], p);
}

// ---------------- weighted message scatter ---------------------------------
// One thread per (edge, 4-channel group): one b128 gather + 4 f32 atomics.
template <int H, int C>
__global__ void edge_scatter(const int* __restrict__ src,
                             const int* __restrict__ dst,
                             const float* __restrict__ xs,
                             const float* __restrict__ p,     // [E,H] numerators
                             const float* __restrict__ den,   // [N,H]
                             float* __restrict__ accum,       // [N,H*C]
                             int nE) {
  constexpr int HC = H * C;
  constexpr int Q = HC / 4;
  int idx = blockIdx.x * blockDim.x + threadIdx.x;
  if (idx >= nE * Q) return;
  int e = idx / Q, q = idx - e * Q;
  int h = (q * 4) / C;
  int s = src[e], d = dst[e];
  float alpha = p[e * H + h] / den[d * H + h];
  float4 v = *(const float4*)(xs + (size_t)s * HC + q * 4);
  float* dp = accum + (size_t)d * HC + q * 4;
  atomicAdd(dp + 0, v.x * alpha);
  atomicAdd(dp + 1, v.y * alpha);
  atomicAdd(dp + 2, v.z * alpha);
  atomicAdd(dp + 3, v.w * alpha);
}

// ---------------- bias + optional ELU into next-layer activations ----------
template <int ELU, int D>
__global__ void finalize_kernel(const float* __restrict__ acc,
                                const float* __restrict__ bias,
                                float* __restrict__ hout, int total4) {
  int i4 = blockIdx.x * blockDim.x + threadIdx.x;
  if (i4 >= total4) return;
  int cbase = (i4 * 4) % D;
  float4 a = ((const float4*)acc)[i4];
  float4 o;
  o.x = a.x + bias[cbase + 0];
  o.y = a.y + bias[cbase + 1];
  o.z = a.z + bias[cbase + 2];
  o.w = a.w + bias[cbase + 3];
  if (ELU) {
    o.x = o.x > 0.f ? o.x : expm1f(o.x);
    o.y = o.y > 0.f ? o.y : expm1f(o.y);
    o.z = o.z > 0.f ? o.z : expm1f(o.z);
    o.w = o.w > 0.f ? o.w : expm1f(o.w);
  }
  ((float4*)hout)[i4] = o;
}

// ---------------- global mean pool -----------------------------------------
__global__ void pool_sum(const float* __restrict__ h, const int* __restrict__ batch,
                         float* __restrict__ pool, float* __restrict__ cnt, int nN) {
  int idx = blockIdx.x * blockDim.x + threadIdx.x;
  if (idx >= nN * HID) return;
  int n = idx / HID, c = idx - n * HID;
  int g = batch[n];
  atomicAdd(&pool[g * HID + c], h[idx]);
  if (c == 0) atomicAdd(&cnt[g], 1.0f);
}

__global__ void pool_mean(const float* __restrict__ pool, const float* __restrict__ cnt,
                          float* __restrict__ emb, int nG) {
  int idx = blockIdx.x * blockDim.x + threadIdx.x;
  if (idx >= nG * HID) return;
  int g = idx / HID;
  float c = cnt[g];
  c = c > 1.0f ? c : 1.0f;
  emb[idx] = pool[idx] / c;
}

// ---------------- one full GATv2 layer -------------------------------------
template <int DIN, int H, int C, int ELU_OUT>
static void run_gat_layer(const float* hin,
                          const float* Wsrc, const float* Wdst, const float* Wedge,
                          const float* bsrc, const float* bdst,
                          const float* att, const float* bias,
                          const int* esrc, const int* edst, const float* eattr,
                          float* xs, float* xd, float* elog, float* mx, float* den,
                          float* accum, float* hout, hipStream_t stream) {
  constexpr int HC = H * C;
  int tiles = (N_NODES / 16) * (HC / 16);
  int gemmBlocks = cdiv(tiles, TPB / 32);
  wmma_gemm_f32<DIN, HC, 0><<<gemmBlocks, TPB, 0, stream>>>(hin, Wsrc, bsrc, xs, N_NODES);
  wmma_gemm_f32<DIN, HC, 0><<<gemmBlocks, TPB, 0, stream>>>(hin, Wdst, bdst, xd, N_NODES);

  fill_kernel<<<cdiv(N_NODES * H, TPB), TPB, 0, stream>>>(mx, -3.402823466e38f, N_NODES * H);
  fill_kernel<<<cdiv(N_NODES * H, TPB), TPB, 0, stream>>>(den, 0.0f, N_NODES * H);
  fill_kernel<<<cdiv(N_NODES * HC, TPB), TPB, 0, stream>>>(accum, 0.0f, N_NODES * HC);

  edge_logits<H, C><<<cdiv(N_EDGES * H, TPB), TPB, 0, stream>>>(
      xs, xd, esrc, edst, eattr, Wedge, att, elog, mx, N_EDGES);
  edge_softmax_num<H><<<cdiv(N_EDGES * H, TPB), TPB, 0, stream>>>(
      edst, elog, mx, den, N_EDGES);
  edge_scatter<H, C><<<cdiv(N_EDGES * (HC / 4), TPB), TPB, 0, stream>>>(
      esrc, edst, xs, elog, den, accum, N_EDGES);
  finalize_kernel<ELU_OUT, HC><<<cdiv(N_NODES * HC / 4, TPB), TPB, 0, stream>>>(
      accum, bias, hout, N_NODES * HC / 4);
}

// ---------------- entry ----------------------------------------------------
extern "C" void kernel_launch(void* const* d_in, const int* in_sizes, int n_in,
                              void* d_out, int out_size, void* d_ws, size_t ws_size,
                              hipStream_t stream) {
  (void)in_sizes; (void)n_in; (void)out_size; (void)ws_size;

  const float* x     = (const float*)d_in[0];
  const int*   esrc  = (const int*)  d_in[1];
  const int*   edst  = (const int*)  d_in[2];
  const float* eattr = (const float*)d_in[3];
  const int*   batch = (const int*)  d_in[4];
  const float* W1s = (const float*)d_in[5];
  const float* W1d = (const float*)d_in[6];
  const float* W1e = (const float*)d_in[7];
  const float* b1s = (const float*)d_in[8];
  const float* b1d = (const float*)d_in[9];
  const float* att1  = (const float*)d_in[10];
  const float* bias1 = (const float*)d_in[11];
  const float* W2s = (const float*)d_in[12];
  const float* W2d = (const float*)d_in[13];
  const float* W2e = (const float*)d_in[14];
  const float* b2s = (const float*)d_in[15];
  const float* b2d = (const float*)d_in[16];
  const float* att2  = (const float*)d_in[17];
  const float* bias2 = (const float*)d_in[18];
  const float* W3s = (const float*)d_in[19];
  const float* W3d = (const float*)d_in[20];
  const float* W3e = (const float*)d_in[21];
  const float* b3s = (const float*)d_in[22];
  const float* b3d = (const float*)d_in[23];
  const float* att3  = (const float*)d_in[24];
  const float* bias3 = (const float*)d_in[25];
  const float* Wm1 = (const float*)d_in[26];
  const float* bm1 = (const float*)d_in[27];
  const float* Wm2 = (const float*)d_in[28];
  const float* bm2 = (const float*)d_in[29];

  // -------- workspace carve-up (floats); node arrays are L2-resident ------
  float* ws   = (float*)d_ws;
  float* hbuf = ws;                                    // [N,128]
  float* xs   = hbuf + (size_t)N_NODES * D2;           // [N,128]
  float* xd   = xs   + (size_t)N_NODES * D2;           // [N,128]
  float* accu = xd   + (size_t)N_NODES * D2;           // [N,128]
  float* elog = accu + (size_t)N_NODES * D2;           // [E,4]
  float* mx   = elog + (size_t)N_EDGES * HEADS;        // [N,4]
  float* den  = mx   + (size_t)N_NODES * HEADS;        // [N,4]
  float* pool = den  + (size_t)N_NODES * HEADS;        // [G,32]
  float* cnt  = pool + (size_t)N_GRAPHS * HID;         // [G]
  float* emb  = cnt  + (size_t)N_GRAPHS;               // [G,32]
  float* t1   = emb  + (size_t)N_GRAPHS * HID;         // [G,64]

  // -------- layer 1: in=16 -> 4 heads x 32, ELU ---------------------------
  run_gat_layer<IN_DIM, HEADS, HID, 1>(x, W1s, W1d, W1e, b1s, b1d, att1, bias1,
                                       esrc, edst, eattr, xs, xd, elog, mx, den,
                                       accu, hbuf, stream);
  // -------- layer 2: in=128 -> 4 heads x 32, ELU --------------------------
  run_gat_layer<D2, HEADS, HID, 1>(hbuf, W2s, W2d, W2e, b2s, b2d, att2, bias2,
                                   esrc, edst, eattr, xs, xd, elog, mx, den,
                                   accu, hbuf, stream);
  // -------- layer 3: in=128 -> 1 head x 32, no ELU ------------------------
  run_gat_layer<D2, 1, HID, 0>(hbuf, W3s, W3d, W3e, b3s, b3d, att3, bias3,
                               esrc, edst, eattr, xs, xd, elog, mx, den,
                               accu, hbuf, stream);

  // -------- global mean pool ----------------------------------------------
  fill_kernel<<<cdiv(N_GRAPHS * HID, TPB), TPB, 0, stream>>>(pool, 0.0f, N_GRAPHS * HID);
  fill_kernel<<<cdiv(N_GRAPHS, TPB), TPB, 0, stream>>>(cnt, 0.0f, N_GRAPHS);
  pool_sum<<<cdiv(N_NODES * HID, TPB), TPB, 0, stream>>>(hbuf, batch, pool, cnt, N_NODES);
  pool_mean<<<cdiv(N_GRAPHS * HID, TPB), TPB, 0, stream>>>(pool, cnt, emb, N_GRAPHS);

  // -------- MLP head (WMMA, relu fused into first GEMM) -------------------
  {
    int tiles1 = (N_GRAPHS / 16) * ((2 * HID) / 16);   // 32 * 4 = 128
    wmma_gemm_f32<HID, 2 * HID, 1><<<cdiv(tiles1, TPB / 32), TPB, 0, stream>>>(
        emb, Wm1, bm1, t1, N_GRAPHS);
    int tiles2 = (N_GRAPHS / 16) * (OUT_DIM / 16);     // 32 * 4 = 128
    wmma_gemm_f32<2 * HID, OUT_DIM, 0><<<cdiv(tiles2, TPB / 32), TPB, 0, stream>>>(
        t1, Wm2, bm2, (float*)d_out, N_GRAPHS);
  }
}